// Mamba2Block_58342835749249
// MI455X (gfx1250) — compile-verified
//
#include <hip/hip_runtime.h>
#include <math.h>

// ---------------- problem constants (match reference) ----------------
#define DIMX   1024
#define DSTATE 64
#define NH     8
#define DI     2048            // D_INNER
#define HD     256             // HEAD_DIM
#define DP     4232            // D_IN_PROJ = 2*DI + NH + 2*DSTATE
#define SEQ    1024
#define BATCH  2
#define BOFF   (2*DI + NH)          // B_proj (=Cm slice) columns [4104,4168)
#define COFF   (2*DI + NH + DSTATE) // C_proj (=dtm slice) columns [4168,4232)

typedef float v2f __attribute__((ext_vector_type(2)));
typedef float v8f __attribute__((ext_vector_type(8)));
typedef int   i32x4 __attribute__((vector_size(4 * sizeof(int))));  // matches builtin proto

// gfx1250 async global->LDS copies (ASYNCcnt-tracked). Guarded so the code
// still compiles (sync fallback) if this toolchain doesn't expose them.
#if __has_builtin(__builtin_amdgcn_global_load_async_to_lds_b128)
#define ASYNC_B128 1
#endif
#if __has_builtin(__builtin_amdgcn_global_load_async_to_lds_b32)
#define ASYNC_B32 1
#endif

__device__ __forceinline__ void async_wait0()
{
#if defined(ASYNC_B128) || defined(ASYNC_B32)
#if __has_builtin(__builtin_amdgcn_s_wait_asynccnt)
    __builtin_amdgcn_s_wait_asynccnt(0);
#else
    asm volatile("s_wait_asynccnt 0x0" ::: "memory");
#endif
#endif
}

// =====================================================================
// GEMM: C[M,N] = A[M,K] * B[N,K]^T   (both row-major, fp32)
// WG = 256 thr (8 waves). WG tile 64x128. Wave tile 16x64 (4 wmma accs).
// K staged through double-buffered LDS in 32-wide chunks; chunk k+1 is
// fetched with global_load_async_to_lds_b128 while chunk k feeds
// V_WMMA_F32_16X16X4_F32. One barrier per chunk.
// =====================================================================
#define KB  32
#define LDA 36   // padded LDS row stride (floats): 16B-aligned, conflict-free

__global__ __launch_bounds__(256)
void gemm_f32_wmma(const float* __restrict__ A, const float* __restrict__ Bw,
                   float* __restrict__ C, int M, int N, int K)
{
    __shared__ float As[2][64  * LDA];
    __shared__ float Bs[2][128 * LDA];

    const int tid  = threadIdx.x;
    const int lane = tid & 31;
    const int w    = tid >> 5;
    const int lm   = lane & 15;      // fragment row/col within 16
    const int hi   = lane >> 4;      // half-wave select (K pair / M+8)

    const int mBase = blockIdx.x * 64;
    const int nBase = blockIdx.y * 128;
    const int wm0   = (w & 3) * 16;  // wave M offset in WG tile
    const int wn0   = (w >> 2) * 64; // wave N offset in WG tile
    const bool fullN = (nBase + 128 <= N);   // uniform per block

    v8f acc[4];
    #pragma unroll
    for (int t = 0; t < 4; ++t) acc[t] = (v8f){0.f,0.f,0.f,0.f,0.f,0.f,0.f,0.f};

    // stage one 64x32 A chunk + 128x32 B chunk at K-offset k0 into buffer buf
    auto stage = [&](int k0, int buf) {
        #pragma unroll
        for (int i = 0; i < 2; ++i) {
            const int e = (i*256 + tid) * 4;          // [0,2048)
            const int r = e >> 5, c = e & 31;
            const float* g = A + (size_t)(mBase + r) * K + k0 + c;
            float*       l = &As[buf][r*LDA + c];
#ifdef ASYNC_B128
            __builtin_amdgcn_global_load_async_to_lds_b128((i32x4*)g, (i32x4*)l, 0, 0);
#else
            *(float4*)l = *(const float4*)g;
#endif
        }
        #pragma unroll
        for (int i = 0; i < 4; ++i) {
            const int e = (i*256 + tid) * 4;          // [0,4096)
            const int r = e >> 5, c = e & 31;
            const int n = nBase + r;
            const float* g = Bw + (size_t)n * K + k0 + c;
            float*       l = &Bs[buf][r*LDA + c];
#ifdef ASYNC_B128
            if (fullN) {
                __builtin_amdgcn_global_load_async_to_lds_b128((i32x4*)g, (i32x4*)l, 0, 0);
            } else
#endif
            {   // ragged-N tile: out-of-range rows must be ZERO in LDS
                float4 v = {0.f, 0.f, 0.f, 0.f};
                if (n < N) v = *(const float4*)g;
                *(float4*)l = v;
            }
        }
    };

    stage(0, 0);
    async_wait0();
    __syncthreads();

    int cur = 0;
    for (int k0 = 0; k0 < K; k0 += KB) {
        if (k0 + KB < K) stage(k0 + KB, cur ^ 1);   // overlap with compute below

        const float* as = As[cur];
        const float* bs = Bs[cur];
        // ---- 32 WMMAs per chunk: A frag reused across 4 N-tiles ----
        #pragma unroll
        for (int kk = 0; kk < KB; kk += 4) {
            const int ak = kk + 2*hi;   // ISA f32 A-layout: v0=K{0|2}, v1=K{1|3}
            const v2f a  = *(const v2f*)(&as[(wm0 + lm)*LDA + ak]);
            const v2f b0 = *(const v2f*)(&bs[(wn0 +  0 + lm)*LDA + ak]);
            const v2f b1 = *(const v2f*)(&bs[(wn0 + 16 + lm)*LDA + ak]);
            const v2f b2 = *(const v2f*)(&bs[(wn0 + 32 + lm)*LDA + ak]);
            const v2f b3 = *(const v2f*)(&bs[(wn0 + 48 + lm)*LDA + ak]);
            acc[0] = __builtin_amdgcn_wmma_f32_16x16x4_f32(false, a, false, b0, (short)0, acc[0], false, false);
            acc[1] = __builtin_amdgcn_wmma_f32_16x16x4_f32(false, a, false, b1, (short)0, acc[1], false, false);
            acc[2] = __builtin_amdgcn_wmma_f32_16x16x4_f32(false, a, false, b2, (short)0, acc[2], false, false);
            acc[3] = __builtin_amdgcn_wmma_f32_16x16x4_f32(false, a, false, b3, (short)0, acc[3], false, false);
        }
        async_wait0();      // next chunk landed in the other LDS buffer
        __syncthreads();
        cur ^= 1;
    }

    // ---- store D: VGPR r = row r (lanes 0-15) / row r+8 (lanes 16-31) ----
    const int mRow = mBase + wm0 + 8*hi;
    #pragma unroll
    for (int t = 0; t < 4; ++t) {
        const int n = nBase + wn0 + 16*t + lm;
        if (n < N) {
            #pragma unroll
            for (int r = 0; r < 8; ++r)
                C[(size_t)(mRow + r) * N + n] = acc[t][r];
        }
    }
}

// =====================================================================
// a_bar[b,t,h] = exp(softplus(Bm + dt_bias[h]) * (-exp(A_log[h])))
// =====================================================================
__global__ __launch_bounds__(256)
void abar_kernel(const float* __restrict__ P, const float* __restrict__ dt_bias,
                 const float* __restrict__ A_log, float* __restrict__ abar)
{
    const int gid = blockIdx.x * 256 + threadIdx.x;   // BATCH*SEQ*NH = 16384
    const int h   = gid & (NH - 1);
    const int row = gid >> 3;
    const float r  = P[(size_t)row * DP + 2*DI + h] + dt_bias[h];
    const float dt = (r > 20.f) ? r : log1pf(__expf(r));
    const float a  = -__expf(A_log[h]);
    abar[gid] = __expf(dt * a);
}

// =====================================================================
// depthwise causal conv(4) + SiLU over the xc slice of zxBCdt
// =====================================================================
__global__ __launch_bounds__(256)
void conv_silu_kernel(const float* __restrict__ P, const float* __restrict__ conv_w,
                      const float* __restrict__ conv_b, float* __restrict__ xh)
{
    const int gid = blockIdx.x * 256 + threadIdx.x;   // BATCH*SEQ*DI
    const int c   = gid & (DI - 1);
    const int row = gid >> 11;
    const int t   = row & (SEQ - 1);
    const float4 w = *(const float4*)(conv_w + c * 4);
    const float* xin = P + (size_t)row * DP + DI + c;
    float acc = conv_b[c];
    if (t >= 3) {
        acc = fmaf(xin[-3*DP], w.x, acc);
        acc = fmaf(xin[-2*DP], w.y, acc);
        acc = fmaf(xin[-1*DP], w.z, acc);
        acc = fmaf(xin[ 0   ], w.w, acc);
    } else {
        const float wk[4] = {w.x, w.y, w.z, w.w};
        #pragma unroll
        for (int k = 0; k < 4; ++k) {
            const int tt = t - 3 + k;
            if (tt >= 0) acc = fmaf(xin[(k - 3) * DP], wk[k], acc);
        }
    }
    xh[gid] = acc / (1.f + __expf(-acc));   // SiLU
}

// =====================================================================
// Selective scan. One WG per (b,h); lane d owns h-state row st[64] in
// VGPRs. B_t/C_t double-buffered in LDS; the next timestep's vectors
// stream in via async global->LDS while the 64-deep FMA chain runs.
// One barrier per timestep.
// =====================================================================
__global__ __launch_bounds__(256)
void scan_kernel(const float* __restrict__ P, const float* __restrict__ xh,
                 const float* __restrict__ abar, const float* __restrict__ Dp_,
                 float* __restrict__ y)
{
    __shared__ float Bs[2][DSTATE];
    __shared__ float Cs[2][DSTATE];
    const int tid = threadIdx.x;                 // d in [0,256)
    const int b   = blockIdx.x >> 3;
    const int h   = blockIdx.x & (NH - 1);
    const float Dp = Dp_[h];

    float st[DSTATE];
    #pragma unroll
    for (int n = 0; n < DSTATE; ++n) st[n] = 0.f;

    // preload t = 0 (synchronous is fine for the prologue)
    if (tid < DSTATE)            Bs[0][tid]        = P[(size_t)(b*SEQ)*DP + BOFF + tid];
    else if (tid < 2*DSTATE)     Cs[0][tid-DSTATE] = P[(size_t)(b*SEQ)*DP + COFF + (tid-DSTATE)];
    __syncthreads();

    for (int t = 0; t < SEQ; ++t) {
        const int cur = t & 1, nxt = cur ^ 1;
        if (t + 1 < SEQ) {   // overlap next-step B/C fetch with the FMA chain
#ifdef ASYNC_B32
            if (tid < DSTATE)
                __builtin_amdgcn_global_load_async_to_lds_b32(
                    (int*)(P + (size_t)(b*SEQ + t + 1)*DP + BOFF + tid),
                    (int*)&Bs[nxt][tid], 0, 0);
            else if (tid < 2*DSTATE)
                __builtin_amdgcn_global_load_async_to_lds_b32(
                    (int*)(P + (size_t)(b*SEQ + t + 1)*DP + COFF + (tid-DSTATE)),
                    (int*)&Cs[nxt][tid-DSTATE], 0, 0);
#else
            if (tid < DSTATE)
                Bs[nxt][tid] = P[(size_t)(b*SEQ + t + 1)*DP + BOFF + tid];
            else if (tid < 2*DSTATE)
                Cs[nxt][tid-DSTATE] = P[(size_t)(b*SEQ + t + 1)*DP + COFF + (tid-DSTATE)];
#endif
        }
        const float a = abar[(b*SEQ + t)*NH + h];
        const float x = xh[((size_t)(b*SEQ + t))*DI + h*HD + tid];
        float acc = 0.f;
        #pragma unroll
        for (int n = 0; n < DSTATE; ++n) {
            st[n] = fmaf(a, st[n], Bs[cur][n] * x);
            acc   = fmaf(st[n], Cs[cur][n], acc);
        }
        y[((size_t)(b*SEQ + t))*DI + h*HD + tid] = fmaf(x, Dp, acc);
        async_wait0();
        __syncthreads();
    }
}

// =====================================================================
// y = RMSNorm(y * SiLU(z)) * norm_w   (in place; one WG per (b,t) row)
// =====================================================================
__global__ __launch_bounds__(256)
void gate_rms_kernel(float* __restrict__ y, const float* __restrict__ P,
                     const float* __restrict__ norm_w)
{
    __shared__ float red[256];
    const int row = blockIdx.x;
    const int tid = threadIdx.x;
    float g[8];
    float ss = 0.f;
    #pragma unroll
    for (int j = 0; j < 8; ++j) {
        const int i = tid + j*256;
        const float z = P[(size_t)row * DP + i];           // z slice = cols [0,DI)
        const float s = z / (1.f + __expf(-z));
        const float v = y[(size_t)row * DI + i] * s;
        g[j] = v;
        ss = fmaf(v, v, ss);
    }
    red[tid] = ss;
    __syncthreads();
    for (int off = 128; off > 0; off >>= 1) {
        if (tid < off) red[tid] += red[tid + off];
        __syncthreads();
    }
    const float inv = rsqrtf(red[0] / (float)DI + 1e-6f);
    #pragma unroll
    for (int j = 0; j < 8; ++j) {
        const int i = tid + j*256;
        y[(size_t)row * DI + i] = g[j] * inv * norm_w[i];
    }
}

// =====================================================================
extern "C" void kernel_launch(void* const* d_in, const int* in_sizes, int n_in,
                              void* d_out, int out_size, void* d_ws, size_t ws_size,
                              hipStream_t stream)
{
    const float* x         = (const float*)d_in[0];
    const float* in_proj_w = (const float*)d_in[1];
    const float* conv_w    = (const float*)d_in[2];
    const float* conv_b    = (const float*)d_in[3];
    const float* A_log     = (const float*)d_in[4];
    const float* D_param   = (const float*)d_in[5];
    const float* dt_bias   = (const float*)d_in[6];
    const float* norm_w    = (const float*)d_in[7];
    const float* out_proj_w= (const float*)d_in[8];
    float* out = (float*)d_out;

    // workspace layout (floats): zxBCdt | xh | abar | y   (~68.3 MB total)
    float* zx   = (float*)d_ws;
    float* xh   = zx   + (size_t)BATCH * SEQ * DP;
    float* abar = xh   + (size_t)BATCH * SEQ * DI;
    float* yb   = abar + (size_t)BATCH * SEQ * NH;

    const int M = BATCH * SEQ;   // 2048

    // 1) in_proj:  zx[M,4232] = x[M,1024] * in_proj_w[4232,1024]^T
    gemm_f32_wmma<<<dim3(M/64, (DP + 127)/128), 256, 0, stream>>>(x, in_proj_w, zx, M, DP, DIMX);
    // 2) a_bar
    abar_kernel<<<(M * NH)/256, 256, 0, stream>>>(zx, dt_bias, A_log, abar);
    // 3) conv + SiLU
    conv_silu_kernel<<<((size_t)M * DI)/256, 256, 0, stream>>>(zx, conv_w, conv_b, xh);
    // 4) selective scan
    scan_kernel<<<BATCH * NH, 256, 0, stream>>>(zx, xh, abar, D_param, yb);
    // 5) gate + RMSNorm (in place on yb)
    gate_rms_kernel<<<M, 256, 0, stream>>>(yb, zx, norm_w);
    // 6) out_proj:  out[M,1024] = yb[M,2048] * out_proj_w[1024,2048]^T
    gemm_f32_wmma<<<dim3(M/64, DIMX/128), 256, 0, stream>>>(yb, out_proj_w, out, M, DIMX, DI);
}